// PartitionRoutingMoE_48361331752980
// MI455X (gfx1250) — compile-verified
//
#include <hip/hip_runtime.h>
#include <math.h>

// ---------------- problem constants ----------------
#define BB 32
#define SS 2048
#define DD 512
#define FF 341
#define EE 6
#define PP 11

#define TM 64            // tokens per workgroup tile (4 M-tiles of 16)
#define XS_STRIDE 516    // 512 + 4 pad (bank-conflict free, 16B-aligned rows)
#define HS_STRIDE 356    // 352 + 4 pad
#define KPAD 344         // F padded to multiple of 4 for GEMM2 K loop

typedef float v2f __attribute__((ext_vector_type(2)));
typedef float v8f __attribute__((ext_vector_type(8)));

// ---- CDNA5 async global->LDS DMA (ASYNCcnt) with portable fallback ----
#if __has_builtin(__builtin_amdgcn_global_load_async_to_lds_b128) && \
    __has_builtin(__builtin_amdgcn_s_wait_asynccnt)
#define USE_ASYNC_LDS 1
typedef int v4i_t __attribute__((vector_size(16)));
typedef v4i_t __attribute__((address_space(1))) gv4i_t;   // global b128
typedef v4i_t __attribute__((address_space(3))) lv4i_t;   // LDS b128
static __device__ __forceinline__ gv4i_t* as_global_b128(const void* p) {
    // generic->global: value-preserving
    return (gv4i_t*)(unsigned long long)p;
}
static __device__ __forceinline__ lv4i_t* as_lds_b128(void* p) {
    // flat LDS aperture address: LDS offset == addr[31:0] (ISA 10.2)
    return (lv4i_t*)(unsigned int)(unsigned long long)p;
}
#else
#define USE_ASYNC_LDS 0
#endif

__device__ __constant__ float c_pw[PP][EE] = {
    {6,0,0,0,0,0},{5,1,0,0,0,0},{4,2,0,0,0,0},{4,1,1,0,0,0},{3,3,0,0,0,0},
    {3,2,1,0,0,0},{3,1,1,1,0,0},{2,2,2,0,0,0},{2,2,1,1,0,0},{2,1,1,1,1,0},
    {1,1,1,1,1,1}};

__device__ __forceinline__ v8f vzero8() {
    v8f z = {0.f,0.f,0.f,0.f,0.f,0.f,0.f,0.f};
    return z;
}
__device__ __forceinline__ float gelu_erf(float v) {
    return 0.5f * v * (1.0f + erff(v * 0.70710678118654752f));
}

// ---------------- router: mean-pool, top-2 of 11 partitions, expert weights,
// and the combined bias vector  Sum_e w_e * b2[e,:]  ----------------
__global__ __launch_bounds__(256) void router_kernel(
    const float* __restrict__ x, const float* __restrict__ rw,
    const float* __restrict__ b2, float* __restrict__ ws_w,
    float* __restrict__ ws_bias) {
    const int b = blockIdx.x;
    const int tid = threadIdx.x;
    __shared__ float m[DD];
    __shared__ float lg[16];
    __shared__ float wsh[8];

    for (int d = tid; d < DD; d += 256) {
        float acc = 0.f;
        const float* xp = x + ((size_t)b * SS) * DD + d;
        for (int s = 0; s < SS; ++s) acc += xp[(size_t)s * DD];
        m[d] = acc * (1.0f / (float)SS);
    }
    __syncthreads();
    if (tid < PP) {
        float acc = 0.f;
        const float* r = rw + tid * DD;
        for (int d = 0; d < DD; ++d) acc += m[d] * r[d];
        lg[tid] = acc;
    }
    __syncthreads();
    if (tid == 0) {
        // top-2 of softmax == top-2 of logits; renormalized pair
        int i0 = 0;
        for (int p = 1; p < PP; ++p) if (lg[p] > lg[i0]) i0 = p;
        int i1 = (i0 == 0) ? 1 : 0;
        for (int p = 0; p < PP; ++p) if (p != i0 && lg[p] > lg[i1]) i1 = p;
        float e1 = __expf(lg[i1] - lg[i0]);
        float v0 = 1.0f / (1.0f + e1);
        float v1 = e1 / (1.0f + e1);
        for (int e = 0; e < EE; ++e) {
            float w = (v0 * c_pw[i0][e] + v1 * c_pw[i1][e]) * (1.0f / 6.0f);
            wsh[e] = (w > 1e-6f) ? w : 0.f;
        }
        wsh[6] = 0.f; wsh[7] = 0.f;
    }
    __syncthreads();
    if (tid < 8) ws_w[b * 8 + tid] = wsh[tid];
    for (int d = tid; d < DD; d += 256) {
        float acc = 0.f;
        for (int e = 0; e < EE; ++e) acc += wsh[e] * b2[e * DD + d];
        ws_bias[b * DD + d] = acc;
    }
}

// ---------------- fused MoE FFN: per 64-token tile, loop active experts,
// GEMM1 (x@W1 -> gelu -> *w_e -> LDS h), GEMM2 (h@W2 -> register acc).
// Waves partition the N dimension only, so every weight element is fetched
// exactly once per block and is amortized over all 4 M-tiles (4 WMMAs/B-frag).
__global__ __launch_bounds__(256, 1) void moe_main_kernel(
    const float* __restrict__ x, const float* __restrict__ W1,
    const float* __restrict__ b1, const float* __restrict__ W2,
    const float* __restrict__ ws_w, const float* __restrict__ ws_bias,
    float* __restrict__ out) {
    extern __shared__ float lds[];
    float* xs = lds;                       // [TM][XS_STRIDE]
    float* hs = lds + TM * XS_STRIDE;      // [TM][HS_STRIDE]

    const int b    = blockIdx.y;
    const int tok0 = blockIdx.x * TM;
    const int tid  = threadIdx.x;
    const int wave = tid >> 5;             // 0..7, owns N-tiles nt % 8 == wave
    const int lane = tid & 31;
    const int l15  = lane & 15;
    const int lhi  = lane >> 4;            // 0: K0/K1 half, 1: K2/K3 half

    // ---- stage x tile into LDS (reused by every expert's GEMM1)
    const float* xg = x + ((size_t)b * SS + tok0) * DD;
#if USE_ASYNC_LDS
    for (int idx = tid; idx < TM * (DD / 4); idx += 256) {
        const int r = idx >> 7, c4 = idx & 127;
        __builtin_amdgcn_global_load_async_to_lds_b128(
            as_global_b128(xg + (size_t)r * DD + c4 * 4),
            as_lds_b128(xs + r * XS_STRIDE + c4 * 4), 0, 0);
    }
#else
    for (int idx = tid; idx < TM * (DD / 4); idx += 256) {
        const int r = idx >> 7, c4 = idx & 127;
        const float4 v = *(const float4*)(xg + (size_t)r * DD + c4 * 4);
        *(float4*)(xs + r * XS_STRIDE + c4 * 4) = v;
    }
#endif
    // overlap: per-batch expert weights while the DMA is in flight
    float wE[EE];
#pragma unroll
    for (int e = 0; e < EE; ++e) wE[e] = ws_w[b * 8 + e];
#if USE_ASYNC_LDS
    __builtin_amdgcn_s_wait_asynccnt(0);
#endif
    __syncthreads();

    // persistent output accumulators: 4 M-tiles x 4 N-tiles of 16x16 f32
    v8f acc[4][4];
#pragma unroll
    for (int m = 0; m < 4; ++m)
#pragma unroll
        for (int j = 0; j < 4; ++j) acc[m][j] = vzero8();

    for (int e = 0; e < EE; ++e) {
        if (wE[e] <= 1e-6f) continue;      // uniform across block: exact skip
        const float* W1e = W1 + (size_t)e * DD * FF;
        const float* W2e = W2 + (size_t)e * FF * DD;
        // warm L2 for this expert's weight panels (global_prefetch_b8)
        __builtin_prefetch(W1e + (size_t)tid * 64, 0, 1);
        __builtin_prefetch(W2e + (size_t)tid * 64, 0, 1);
        __syncthreads();                   // hs no longer read by prev GEMM2

        // ======== GEMM1: h = gelu(x @ W1_e + b1_e) * w_e ========
        for (int nt = wave; nt < 22; nt += 8) {
            const int ncol = nt * 16 + l15;
            const int nc   = (ncol < FF) ? ncol : (FF - 1);
            v8f c[4];
#pragma unroll
            for (int m = 0; m < 4; ++m) c[m] = vzero8();
            for (int k = 0; k < DD; k += 4) {
                const int kk = k + 2 * lhi;
                v2f bf;
                bf.x = W1e[(size_t)kk * FF + nc];
                bf.y = W1e[(size_t)(kk + 1) * FF + nc];
#pragma unroll
                for (int m = 0; m < 4; ++m) {
                    v2f a = *(const v2f*)(xs + (m * 16 + l15) * XS_STRIDE + kk);
                    c[m] = __builtin_amdgcn_wmma_f32_16x16x4_f32(
                               false, a, false, bf, (short)0, c[m],
                               false, false);
                }
            }
            const float bias = (ncol < FF) ? b1[e * FF + ncol] : 0.f;
            const float we = wE[e];
#pragma unroll
            for (int m = 0; m < 4; ++m) {
#pragma unroll
                for (int i = 0; i < 8; ++i) {
                    const int trow = m * 16 + i + 8 * lhi;
                    float hv = 0.f;
                    if (ncol < FF) hv = gelu_erf(c[m][i] + bias) * we;
                    hs[trow * HS_STRIDE + ncol] = hv;  // pad cols >=341 -> 0
                }
            }
        }
        __syncthreads();                   // hs complete

        // ======== GEMM2: acc += h @ W2_e (h already carries w_e) ========
        for (int k = 0; k < KPAD; k += 4) {
            const int kk  = k + 2 * lhi;
            const int kr0 = (kk < FF) ? kk : (FF - 1);       // h rows >=341
            const int kr1 = (kk + 1 < FF) ? kk + 1 : (FF - 1); // are 0: exact
            v2f a[4];
#pragma unroll
            for (int m = 0; m < 4; ++m)
                a[m] = *(const v2f*)(hs + (m * 16 + l15) * HS_STRIDE + kk);
#pragma unroll
            for (int j = 0; j < 4; ++j) {
                const int ncol = (wave + 8 * j) * 16 + l15;
                v2f bf;
                bf.x = W2e[(size_t)kr0 * DD + ncol];
                bf.y = W2e[(size_t)kr1 * DD + ncol];
#pragma unroll
                for (int m = 0; m < 4; ++m)
                    acc[m][j] = __builtin_amdgcn_wmma_f32_16x16x4_f32(
                                    false, a[m], false, bf, (short)0,
                                    acc[m][j], false, false);
            }
        }
    }

    // ---- write out += combined bias  Sum_e w_e*b2[e,:]
    float* outg = out + ((size_t)b * SS + tok0) * DD;
#pragma unroll
    for (int j = 0; j < 4; ++j) {
        const int ncol = (wave + 8 * j) * 16 + l15;
        const float bias = ws_bias[b * DD + ncol];
#pragma unroll
        for (int m = 0; m < 4; ++m) {
#pragma unroll
            for (int i = 0; i < 8; ++i) {
                const int trow = m * 16 + i + 8 * lhi;
                outg[(size_t)trow * DD + ncol] = acc[m][j][i] + bias;
            }
        }
    }
}

extern "C" void kernel_launch(void* const* d_in, const int* in_sizes, int n_in,
                              void* d_out, int out_size, void* d_ws, size_t ws_size,
                              hipStream_t stream) {
    const float* x  = (const float*)d_in[0];
    const float* rw = (const float*)d_in[1];
    const float* W1 = (const float*)d_in[2];
    const float* b1 = (const float*)d_in[3];
    const float* W2 = (const float*)d_in[4];
    const float* b2 = (const float*)d_in[5];
    float* out = (float*)d_out;

    float* ws_w    = (float*)d_ws;        // [32][8] expert weights
    float* ws_bias = ws_w + BB * 8;       // [32][512] combined b2 term

    router_kernel<<<BB, 256, 0, stream>>>(x, rw, b2, ws_w, ws_bias);

    const size_t lds_bytes = (size_t)(TM * XS_STRIDE + TM * HS_STRIDE) * sizeof(float);
    dim3 grid(SS / TM, BB);               // 32 token tiles x 32 batches
    moe_main_kernel<<<grid, 256, lds_bytes, stream>>>(x, W1, b1, W2,
                                                      ws_w, ws_bias, out);
}